// MoEGate_26508538151526
// MI455X (gfx1250) — compile-verified
//
#include <hip/hip_runtime.h>

// MoE gating for MI455X (gfx1250, wave32).
// T=32768 tokens, H=2048, E=8 experts, top-2, seq-aux loss.
// Bandwidth-bound: 256MB of fp32 activations -> ~11us floor @ 23.3TB/s.
// Matrix math via V_WMMA_F32_16X16X4_F32 (full fp32 precision for stable
// top-k ordering), one wave per 16-token tile, K-loop over H with b128
// loads (K-permuted so one float4 feeds two WMMA steps) and no branches
// in the inner loop (expert padding handled via ignored C columns 8..15).

typedef __attribute__((ext_vector_type(2))) float v2f;
typedef __attribute__((ext_vector_type(4))) float v4f;
typedef __attribute__((ext_vector_type(8))) float v8f;

#define T_TOKENS 32768
#define HDIM     2048
#define NEXP     8
#define SEQ      8192
#define NBATCH   4
#define TOPK     2
#define ALPHA_LB 0.1f

// ---------------------------------------------------------------------------
// Kernel 0: zero the 64-float aux accumulator workspace (ce/pi per batch).
// ---------------------------------------------------------------------------
__global__ void ws_zero_kernel(float* __restrict__ ws) {
    if (threadIdx.x < 64) ws[threadIdx.x] = 0.0f;
}

// ---------------------------------------------------------------------------
// Kernel 1: logits via f32 WMMA, softmax, top-2, renorm, aux partials.
// grid = 256 blocks x 256 threads (8 waves); wave g handles tokens
// [16g, 16g+16). Each block covers 128 consecutive tokens (single batch).
// ---------------------------------------------------------------------------
__global__ __launch_bounds__(256) void moe_gate_kernel(
    const float* __restrict__ x,     // [T, H]
    const float* __restrict__ w,     // [E, H]
    float* __restrict__ out,         // [2T idx][2T wt][1 aux]
    float* __restrict__ ws)          // [NBATCH][16] : pi sums (0..7), ce counts (8..15)
{
    __shared__ float s_log[8][16][16];  // per-wave 16x16 logits tile (8KB)
    __shared__ float s_aux[16];         // block aux accum: [0..7]=pi, [8..15]=ce

    const int tid  = threadIdx.x;
    const int lane = tid & 31;
    const int wv   = tid >> 5;                 // wave in block: 0..7
    const int l16  = lane & 15;
    const int half = lane >> 4;                // 0 or 1 -> K sub-quad
    const int tok0 = (blockIdx.x * 8 + wv) * 16;

    if (tid < 16) s_aux[tid] = 0.0f;
    __syncthreads();

    // K-permuted fragment sourcing: step j consumes K-tuples
    //   {8j, 8j+1, 8j+4, 8j+5} (from .xy of the float4)
    //   {8j+2,8j+3, 8j+6, 8j+7} (from .zw)
    // so lane `half` loads one float4 at column 8j + 4*half for both A and B.
    // A: row = token (M = l16). B: column N = l16 holds W row (l16&7);
    // columns 8..15 of C become duplicate logits we simply never read.
    const float* arow = x + (size_t)(tok0 + l16) * HDIM + 4 * half;
    const float* brow = w + (size_t)(l16 & 7)   * HDIM + 4 * half;

    v8f c = {0.f, 0.f, 0.f, 0.f, 0.f, 0.f, 0.f, 0.f};

#pragma unroll 4
    for (int k = 0; k < HDIM; k += 8) {
        // x is streamed exactly once (256MB > 192MB L2): non-temporal.
        v4f a = __builtin_nontemporal_load((const v4f*)(arow + k));
        v4f b = *(const v4f*)(brow + k);     // W stays cache-resident
        v2f a0 = {a.x, a.y}, a1 = {a.z, a.w};
        v2f b0 = {b.x, b.y}, b1 = {b.z, b.w};
        c = __builtin_amdgcn_wmma_f32_16x16x4_f32(
                false, a0, false, b0, (short)0, c, false, false);
        c = __builtin_amdgcn_wmma_f32_16x16x4_f32(
                false, a1, false, b1, (short)0, c, false, false);
    }

    // Spill C to LDS: VGPR r holds token (r + 8*half), expert column l16.
#pragma unroll
    for (int r = 0; r < 8; ++r) {
        s_log[wv][r + 8 * half][l16] = c[r];
    }
    // Same-wave LDS store->load is in order (DScnt); no barrier needed here.

    if (lane < 16) {
        const int t = tok0 + lane;
        float v[NEXP];
        float mx = -3.4e38f;
#pragma unroll
        for (int e = 0; e < NEXP; ++e) {
            v[e] = s_log[wv][lane][e];
            mx = fmaxf(mx, v[e]);
        }
        float sum = 0.f;
#pragma unroll
        for (int e = 0; e < NEXP; ++e) {
            v[e] = __expf(v[e] - mx);
            sum += v[e];
        }
        const float inv = 1.0f / sum;
#pragma unroll
        for (int e = 0; e < NEXP; ++e) v[e] *= inv;   // softmax scores

        // top-2 (ties -> lowest index, matching jax.lax.top_k)
        int i0 = 0; float p0 = v[0];
#pragma unroll
        for (int e = 1; e < NEXP; ++e)
            if (v[e] > p0) { p0 = v[e]; i0 = e; }
        int i1 = -1; float p1 = -1.0f;
#pragma unroll
        for (int e = 0; e < NEXP; ++e)
            if (e != i0 && v[e] > p1) { p1 = v[e]; i1 = e; }

        const float dn = 1.0f / (p0 + p1 + 1e-20f);
        out[2 * t + 0]                = (float)i0;
        out[2 * t + 1]                = (float)i1;
        out[2 * T_TOKENS + 2 * t + 0] = p0 * dn;
        out[2 * T_TOKENS + 2 * t + 1] = p1 * dn;

        // aux-loss partials into block LDS (ds_add_f32 atomics)
#pragma unroll
        for (int e = 0; e < NEXP; ++e) atomicAdd(&s_aux[e], v[e]);
        atomicAdd(&s_aux[8 + i0], 1.0f);
        atomicAdd(&s_aux[8 + i1], 1.0f);
    }

    __syncthreads();
    if (tid < 16) {
        const int b = (blockIdx.x * 128) / SEQ;   // constant per block
        atomicAdd(&ws[b * 16 + tid], s_aux[tid]);
    }
}

// ---------------------------------------------------------------------------
// Kernel 2: finalize scalar aux loss from the 64-float workspace.
// aux = ALPHA * mean_b( sum_e (ce[b][e] / (S*K/E)) * (pi_sum[b][e] / S) )
// ---------------------------------------------------------------------------
__global__ void moe_aux_finalize_kernel(const float* __restrict__ ws,
                                        float* __restrict__ out) {
    if (threadIdx.x == 0 && blockIdx.x == 0) {
        const float ce_norm = (float)NEXP / ((float)SEQ * (float)TOPK);
        const float pi_norm = 1.0f / (float)SEQ;
        float total = 0.f;
        for (int b = 0; b < NBATCH; ++b) {
            float acc = 0.f;
            for (int e = 0; e < NEXP; ++e) {
                const float pi = ws[b * 16 + e] * pi_norm;
                const float ce = ws[b * 16 + 8 + e] * ce_norm;
                acc += ce * pi;
            }
            total += acc;
        }
        out[4 * T_TOKENS] = ALPHA_LB * (total / (float)NBATCH);
    }
}

// ---------------------------------------------------------------------------
extern "C" void kernel_launch(void* const* d_in, const int* in_sizes, int n_in,
                              void* d_out, int out_size, void* d_ws, size_t ws_size,
                              hipStream_t stream) {
    const float* x = (const float*)d_in[0];   // [4,8192,2048] f32
    const float* w = (const float*)d_in[1];   // [8,2048] f32
    float* out = (float*)d_out;
    float* ws  = (float*)d_ws;

    hipLaunchKernelGGL(ws_zero_kernel, dim3(1), dim3(64), 0, stream, ws);
    hipLaunchKernelGGL(moe_gate_kernel, dim3(256), dim3(256), 0, stream,
                       x, w, out, ws);
    hipLaunchKernelGGL(moe_aux_finalize_kernel, dim3(1), dim3(32), 0, stream,
                       ws, out);
}